// Learned2DRelativeSelfAttention_59863254171748
// MI455X (gfx1250) — compile-verified
//
#include <hip/hip_runtime.h>
#include <math.h>

typedef float v2f __attribute__((ext_vector_type(2)));
typedef float v8f __attribute__((ext_vector_type(8)));

#define NUM_HEADS 9
#define HIDDEN    256
#define HW        32        // h == w == 32
#define NB        32        // batch
#define POSD      128

// workspace layout (float offsets)
#define WS_PW 0
#define WS_PH (WS_PW + NUM_HEADS*HW*HW)                 // 9216
#define WS_T  (WS_PH + NUM_HEADS*HW*HW)                 // 18432
#define WS_IV (WS_T  + NB*HW*HW*HIDDEN)                 // +8388608
#define WS_WT (WS_IV + NB*HW*HW*HIDDEN)                 // +8388608
#define WS_VT (WS_WT + NUM_HEADS*HIDDEN*HIDDEN)         // +589824
// total floats = WS_VT + 8388608 = 25,774,080  (~103 MB)

// ---------------------------------------------------------------------------
// pw[n][j][l] = softmax_l( dot(width_emb[l-j+31], hk_w[n]) ) * (1/16)
// ph[n][i][k] = softmax_k( dot(height_emb[k-i+31], hk_h[n]) )
// one wave32 per (dir, n, row)
__global__ void prep_softmax(const float* __restrict__ wemb,
                             const float* __restrict__ hemb,
                             const float* __restrict__ hkw,
                             const float* __restrict__ hkh,
                             float* __restrict__ ws) {
  int dir = blockIdx.x / (NUM_HEADS*HW);
  int rem = blockIdx.x % (NUM_HEADS*HW);
  int n = rem / HW, q = rem % HW;
  int l = threadIdx.x;                       // 0..31 (wave32)
  const float* emb = dir ? hemb : wemb;
  const float* hk  = dir ? hkh  : hkw;
  float* outp = ws + (dir ? WS_PH : WS_PW);

  int r = l - q + (HW - 1);                  // 0..62
  const float* e = emb + r*POSD;
  const float* k = hk  + n*POSD;
  float s = 0.f;
  for (int p = 0; p < POSD; ++p) s += e[p]*k[p];

  float m = s;
  for (int off = 16; off >= 1; off >>= 1) m = fmaxf(m, __shfl_xor(m, off, 32));
  float ex = expf(s - m);
  float sum = ex;
  for (int off = 16; off >= 1; off >>= 1) sum += __shfl_xor(sum, off, 32);
  float p = ex / sum;
  if (dir == 0) p *= 0.0625f;                // fold 1/sqrt(HIDDEN)
  outp[(n*HW + q)*HW + l] = p;
}

// vT[((b*32+k)*32+l)*256 + d] = hidden[((b*256+d)*32+k)*32 + l]
__global__ void transpose_v(const float* __restrict__ hid, float* __restrict__ vT) {
  int idx = blockIdx.x*256 + threadIdx.x;    // 8,388,608 total
  int d = idx & 255, l = (idx >> 8) & 31, k = (idx >> 13) & 31, b = idx >> 18;
  vT[idx] = hid[((b*HIDDEN + d)*HW + k)*HW + l];
}

// WT[n][d][c] = fv_W[c][n*256 + d]
__global__ void transpose_w(const float* __restrict__ fvW, float* __restrict__ WT) {
  int idx = blockIdx.x*256 + threadIdx.x;    // 589,824 total
  int c = idx & 255, d = (idx >> 8) & 255, n = idx >> 16;
  WT[idx] = fvW[c*(NUM_HEADS*HIDDEN) + n*HIDDEN + d];
}

// out[b][c][i][j] = fv_b[c]  (accumulation target for gemm3)
__global__ void init_out(const float* __restrict__ fvb, float* __restrict__ out) {
  int idx = blockIdx.x*256 + threadIdx.x;    // 2,097,152 float4 stores
  int base = idx*4;
  float bv = fvb[(base >> 10) & 255];
  float4 v = make_float4(bv, bv, bv, bv);
  *(float4*)(out + base) = v;
}

// probs[i,j,n,k,l] = ph[n,i,k] * pw[n,j,l]   (pw carries the 1/16)
__global__ void write_probs(const float* __restrict__ ws, float* __restrict__ probs) {
  int idx = blockIdx.x*256 + threadIdx.x;    // 2,359,296 float4 stores
  int l4 = idx & 7;
  int t  = idx >> 3;
  int k = t & 31; t >>= 5;
  int n = t % 9;  t /= 9;
  int j = t & 31; int i = t >> 5;
  const float* pw = ws + WS_PW;
  const float* ph = ws + WS_PH;
  float phs = ph[(n*HW + i)*HW + k];
  float4 pw4 = *(const float4*)(pw + (n*HW + j)*HW + l4*4);
  float4 o = make_float4(phs*pw4.x, phs*pw4.y, phs*pw4.z, phs*pw4.w);
  *(float4*)(probs + idx*4) = o;
}

// ---------------------------------------------------------------------------
// GEMM1 per head n:  t[b,k,j,d] = sum_l pw[n,j,l] * vT[b,k,l,d]
// grid (32 b, 32 k, 4 dgrp) x 256 threads; each wave = one 16(j) x 16(d) tile
__global__ void gemm1_tn(int n, const float* __restrict__ vT, float* __restrict__ wsbuf) {
  const float* pw = wsbuf + WS_PW + n*HW*HW;
  float* t = wsbuf + WS_T;
  int b = blockIdx.x, k = blockIdx.y;
  int wave = threadIdx.x >> 5, lane = threadIdx.x & 31;
  int half = lane >> 4, lid = lane & 15;
  int jbase = (wave & 1)*16;
  int dbase = (blockIdx.z*4 + (wave >> 1))*16;
  const float* Arow = pw + (jbase + lid)*HW;                      // A[j][l]
  const float* Bcol = vT + ((b*HW + k)*HW)*HIDDEN + dbase + lid;  // B[l][d], l stride=256
  v8f acc = {};
  for (int kk = 0; kk < HW; kk += 4) {
    v2f a, bb;
    a.x  = Arow[kk + 2*half];
    a.y  = Arow[kk + 2*half + 1];
    bb.x = Bcol[(kk + 2*half)*HIDDEN];
    bb.y = Bcol[(kk + 2*half + 1)*HIDDEN];
    acc = __builtin_amdgcn_wmma_f32_16x16x4_f32(false, a, false, bb,
                                                (short)0, acc, false, false);
  }
  float* Out = t + ((b*HW + k)*HW + jbase)*HIDDEN + dbase + lid;
#pragma unroll
  for (int r = 0; r < 8; ++r) Out[(r + 8*half)*HIDDEN] = acc[r];
}

// GEMM2 per head n:  iv[b,i,j,d] = sum_k ph[n,i,k] * t[b,k,j,d]
__global__ void gemm2_iv(int n, float* __restrict__ wsbuf) {
  const float* ph = wsbuf + WS_PH + n*HW*HW;
  const float* t  = wsbuf + WS_T;
  float* iv = wsbuf + WS_IV;
  int b = blockIdx.x, j = blockIdx.y;
  int wave = threadIdx.x >> 5, lane = threadIdx.x & 31;
  int half = lane >> 4, lid = lane & 15;
  int ibase = (wave & 1)*16;
  int dbase = (blockIdx.z*4 + (wave >> 1))*16;
  const float* Arow = ph + (ibase + lid)*HW;                      // A[i][k]
  const float* Bcol = t + (b*HW*HW + j)*HIDDEN + dbase + lid;     // B[k][d], k stride=8192
  v8f acc = {};
  for (int kk = 0; kk < HW; kk += 4) {
    v2f a, bb;
    a.x  = Arow[kk + 2*half];
    a.y  = Arow[kk + 2*half + 1];
    bb.x = Bcol[(kk + 2*half)*HW*HIDDEN];
    bb.y = Bcol[(kk + 2*half + 1)*HW*HIDDEN];
    acc = __builtin_amdgcn_wmma_f32_16x16x4_f32(false, a, false, bb,
                                                (short)0, acc, false, false);
  }
  float* Out = iv + ((b*HW + ibase)*HW + j)*HIDDEN + dbase + lid;
#pragma unroll
  for (int r = 0; r < 8; ++r) Out[(r + 8*half)*HW*HIDDEN] = acc[r];
}

// GEMM3 per head n:  out[b,c,i,j] += sum_d iv[(b,i,j),d] * WT[n,d,c]
// 32768x256 @ 256x256; each wave one 16x16 tile, C preloaded from out.
__global__ void gemm3_out(int n, const float* __restrict__ wsbuf, float* __restrict__ out) {
  const float* iv = wsbuf + WS_IV;
  const float* WT = wsbuf + WS_WT + n*HIDDEN*HIDDEN;
  int gw = blockIdx.x*8 + (threadIdx.x >> 5);
  int lane = threadIdx.x & 31, half = lane >> 4, lid = lane & 15;
  int mt = gw >> 4, ct = gw & 15;
  int rbase = mt*16, cbase = ct*16;
  const float* Arow = iv + (rbase + lid)*HIDDEN;   // A[row][d]
  const float* Bcol = WT + cbase + lid;            // B[d][c], d stride=256
  v8f acc;
#pragma unroll
  for (int r = 0; r < 8; ++r) {
    int row = rbase + r + 8*half;                  // row = (b,i,j) flat
    acc[r] = out[(row >> 10)*(HIDDEN*HW*HW) + (cbase + lid)*(HW*HW) + (row & 1023)];
  }
  for (int kk = 0; kk < HIDDEN; kk += 4) {
    v2f a, bb;
    a.x  = Arow[kk + 2*half];
    a.y  = Arow[kk + 2*half + 1];
    bb.x = Bcol[(kk + 2*half)*HIDDEN];
    bb.y = Bcol[(kk + 2*half + 1)*HIDDEN];
    acc = __builtin_amdgcn_wmma_f32_16x16x4_f32(false, a, false, bb,
                                                (short)0, acc, false, false);
  }
#pragma unroll
  for (int r = 0; r < 8; ++r) {
    int row = rbase + r + 8*half;
    out[(row >> 10)*(HIDDEN*HW*HW) + (cbase + lid)*(HW*HW) + (row & 1023)] = acc[r];
  }
}

// ---------------------------------------------------------------------------
extern "C" void kernel_launch(void* const* d_in, const int* in_sizes, int n_in,
                              void* d_out, int out_size, void* d_ws, size_t ws_size,
                              hipStream_t stream) {
  (void)in_sizes; (void)n_in; (void)out_size; (void)ws_size;
  const float* hid  = (const float*)d_in[0];
  const float* wemb = (const float*)d_in[1];
  const float* hemb = (const float*)d_in[2];
  const float* hkw  = (const float*)d_in[3];
  const float* hkh  = (const float*)d_in[4];
  const float* fvW  = (const float*)d_in[5];
  const float* fvb  = (const float*)d_in[6];
  float* out   = (float*)d_out;
  float* probs = out + NB*HIDDEN*HW*HW;            // second output, flat concat
  float* ws    = (float*)d_ws;

  prep_softmax<<<2*NUM_HEADS*HW, 32, 0, stream>>>(wemb, hemb, hkw, hkh, ws);
  transpose_v<<<(NB*HW*HW*HIDDEN)/256, 256, 0, stream>>>(hid, ws + WS_VT);
  transpose_w<<<(NUM_HEADS*HIDDEN*HIDDEN)/256, 256, 0, stream>>>(fvW, ws + WS_WT);
  init_out<<<(NB*HIDDEN*HW*HW)/4/256, 256, 0, stream>>>(fvb, out);
  write_probs<<<(HW*HW*NUM_HEADS*HW*HW)/4/256, 256, 0, stream>>>(ws, probs);

  for (int n = 0; n < NUM_HEADS; ++n) {
    gemm1_tn<<<dim3(NB, HW, 4), 256, 0, stream>>>(n, ws + WS_VT, ws);
    gemm2_iv<<<dim3(NB, HW, 4), 256, 0, stream>>>(n, ws);
    gemm3_out<<<4096, 256, 0, stream>>>(n, ws, out);
  }
}